// SmallerNet_63402307224408
// MI455X (gfx1250) — compile-verified
//
#include <hip/hip_runtime.h>
#include <hip/hip_bf16.h>

typedef float v2f __attribute__((ext_vector_type(2)));
typedef float v8f __attribute__((ext_vector_type(8)));

#define N_NODES 10000
#define DIM     256
#define KPAD    516   // LDS row stride in floats: 516 % 64 == 4 -> conflict-free b64 reads

// ---------------------------------------------------------------------------
// Zero a float region (grid-stride).
// ---------------------------------------------------------------------------
__global__ void zero_kernel(float* __restrict__ p, long long n) {
    long long i = (long long)blockIdx.x * blockDim.x + threadIdx.x;
    long long stride = (long long)gridDim.x * blockDim.x;
    for (; i < n; i += stride) p[i] = 0.0f;
}

// ---------------------------------------------------------------------------
// Edge aggregation: one wave per edge; lane handles 8 floats (2x float4).
// agg[dst] += x[src]; deg[dst] += 1. x and agg are L2-resident (~10 MB each).
// ---------------------------------------------------------------------------
__global__ void agg_kernel(const float* __restrict__ x,
                           const long long* __restrict__ ei, long long E,
                           float* __restrict__ agg, float* __restrict__ deg) {
    int lane = threadIdx.x & 31;
    long long wave   = (long long)blockIdx.x * (blockDim.x >> 5) + (threadIdx.x >> 5);
    long long nwaves = (long long)gridDim.x * (blockDim.x >> 5);
    for (long long e = wave; e < E; e += nwaves) {
        long long s = ei[e];
        long long d = ei[E + e];
        const float4* xr = (const float4*)(x + s * DIM);
        float4 v0 = xr[lane];
        float4 v1 = xr[lane + 32];
        float* ar = agg + d * DIM + lane * 4;
        atomicAdd(ar + 0, v0.x);
        atomicAdd(ar + 1, v0.y);
        atomicAdd(ar + 2, v0.z);
        atomicAdd(ar + 3, v0.w);
        atomicAdd(ar + 128 + 0, v1.x);
        atomicAdd(ar + 128 + 1, v1.y);
        atomicAdd(ar + 128 + 2, v1.z);
        atomicAdd(ar + 128 + 3, v1.w);
        if (lane == 0) atomicAdd(deg + d, 1.0f);
    }
}

// ---------------------------------------------------------------------------
// agg[i] /= max(deg[row], 1)
// ---------------------------------------------------------------------------
__global__ void scale_kernel(float* __restrict__ agg, const float* __restrict__ deg,
                             long long n) {
    long long i = (long long)blockIdx.x * blockDim.x + threadIdx.x;
    if (i < n) {
        long long row = i >> 8;  // DIM == 256
        agg[i] *= 1.0f / fmaxf(deg[row], 1.0f);
    }
}

// ---------------------------------------------------------------------------
// Generic fp32 WMMA GEMM:  out[M,N] = act( A1[M,K1]*W1[N,K1]^T
//                                        + A2[M,K2]*W2[N,K2]^T + bias )
// Block: 256 threads = 8 waves. Block tile 128(M) x 16(N); wave tile 16x16.
// Weight tile (16 x K) staged in LDS, padded stride for bank-conflict-free
// ds_load_b64. Uses V_WMMA_F32_16X16X4_F32 (fp32 in/out, matches reference).
// ---------------------------------------------------------------------------
__global__ __launch_bounds__(256) void gemm_wmma_f32(
    const float* __restrict__ A1, int K1,
    const float* __restrict__ A2, int K2,
    const float* __restrict__ W1, const float* __restrict__ W2,
    const float* __restrict__ bias, float* __restrict__ out,
    int M, int N, int ldo, int do_relu) {
    __shared__ float lw[16 * KPAD];
    const int K = K1 + K2;
    const int nBase = blockIdx.y * 16;

    // Cooperative fill of the 16 x K weight tile (zeros for n >= N).
    for (int nr = 0; nr < 16; ++nr) {
        int gn = nBase + nr;
        for (int kk = threadIdx.x; kk < K; kk += 256) {
            float v = 0.0f;
            if (gn < N)
                v = (kk < K1) ? W1[(long long)gn * K1 + kk]
                              : W2[(long long)gn * K2 + (kk - K1)];
            lw[nr * KPAD + kk] = v;
        }
    }
    __syncthreads();

    const int lane  = threadIdx.x & 31;
    const int wave  = threadIdx.x >> 5;
    const int mBase = blockIdx.x * 128 + wave * 16;
    const int nl    = lane & 15;            // N index within tile / M row for A frag
    const int koff  = (lane >> 4) << 1;     // lanes 0-15 -> K{0,1}; 16-31 -> K{2,3}

    int mrow = mBase + nl;                  // clamp (EXEC must stay all-1s for WMMA)
    if (mrow >= M) mrow = M - 1;

    v8f acc = {0.f, 0.f, 0.f, 0.f, 0.f, 0.f, 0.f, 0.f};

    const float* a1row = A1 + (long long)mrow * K1;
    for (int k = 0; k < K1; k += 4) {
        v2f a = *(const v2f*)(a1row + k + koff);
        v2f b = *(const v2f*)(lw + nl * KPAD + k + koff);
        acc = __builtin_amdgcn_wmma_f32_16x16x4_f32(false, a, false, b,
                                                    (short)0, acc, false, false);
    }
    if (K2 > 0) {
        const float* a2row = A2 + (long long)mrow * K2;
        for (int k = 0; k < K2; k += 4) {
            v2f a = *(const v2f*)(a2row + k + koff);
            v2f b = *(const v2f*)(lw + nl * KPAD + K1 + k + koff);
            acc = __builtin_amdgcn_wmma_f32_16x16x4_f32(false, a, false, b,
                                                        (short)0, acc, false, false);
        }
    }

    // Epilogue: C/D layout — lane<16: VGPR j = row j; lane>=16: VGPR j = row 8+j.
    const int gn = nBase + nl;
    const float bv = (gn < N) ? bias[gn] : 0.0f;
    const int mOff = mBase + ((lane >> 4) << 3);
#pragma unroll
    for (int j = 0; j < 8; ++j) {
        int row = mOff + j;
        if (row < M && gn < N) {
            float v = acc[j] + bv;
            if (do_relu) v = fmaxf(v, 0.0f);
            out[(long long)row * ldo + gn] = v;
        }
    }
}

// ---------------------------------------------------------------------------
// Launch
// ---------------------------------------------------------------------------
extern "C" void kernel_launch(void* const* d_in, const int* in_sizes, int n_in,
                              void* d_out, int out_size, void* d_ws, size_t ws_size,
                              hipStream_t stream) {
    const float*     x   = (const float*)d_in[0];
    const long long* ei  = (const long long*)d_in[1];   // int64 edge_index [2,E]
    const float*     W_l = (const float*)d_in[2];
    const float*     b_l = (const float*)d_in[3];
    const float*     W_r = (const float*)d_in[4];
    const float*     Wa  = (const float*)d_in[5];
    const float*     ba  = (const float*)d_in[6];
    const float*     W1  = (const float*)d_in[7];
    const float*     b1  = (const float*)d_in[8];
    const float*     W2  = (const float*)d_in[9];
    const float*     b2  = (const float*)d_in[10];
    const float*     W3  = (const float*)d_in[11];
    const float*     b3  = (const float*)d_in[12];
    float* out = (float*)d_out;

    const long long E = in_sizes[1] / 2;
    const long long ND = (long long)N_NODES * DIM;

    float* agg = (float*)d_ws;
    float* deg = agg + ND;
    float* h1  = deg + N_NODES;                       // 10000 x 256
    float* h2  = h1 + (long long)N_NODES * 256;       // 10000 x 128
    float* h3  = h2 + (long long)N_NODES * 128;       // 10000 x 64
    float* h4  = h3 + (long long)N_NODES * 64;        // 10000 x 32

    // 1) zero accumulators (agg + deg), every call (graph replay determinism)
    zero_kernel<<<1024, 256, 0, stream>>>(agg, ND + N_NODES);
    // 2) scatter-sum over edges
    agg_kernel<<<2048, 256, 0, stream>>>(x, ei, E, agg, deg);
    // 3) mean normalization
    scale_kernel<<<(int)((ND + 255) / 256), 256, 0, stream>>>(agg, deg, ND);

    const int gridM = (N_NODES + 127) / 128;  // 79
    // 4) fused SAGE layer: relu(agg*W_l^T + b_l + x*W_r^T), K = 256+256
    gemm_wmma_f32<<<dim3(gridM, 16), 256, 0, stream>>>(
        agg, 256, x, 256, W_l, W_r, b_l, h1, N_NODES, 256, 256, 1);
    // 5) MLP tower
    gemm_wmma_f32<<<dim3(gridM, 8), 256, 0, stream>>>(
        h1, 256, nullptr, 0, Wa, nullptr, ba, h2, N_NODES, 128, 128, 1);
    gemm_wmma_f32<<<dim3(gridM, 4), 256, 0, stream>>>(
        h2, 128, nullptr, 0, W1, nullptr, b1, h3, N_NODES, 64, 64, 1);
    gemm_wmma_f32<<<dim3(gridM, 2), 256, 0, stream>>>(
        h3, 64, nullptr, 0, W2, nullptr, b2, h4, N_NODES, 32, 32, 1);
    gemm_wmma_f32<<<dim3(gridM, 1), 256, 0, stream>>>(
        h4, 32, nullptr, 0, W3, nullptr, b3, out, N_NODES, 3, 3, 0);
}